// GRCN_60181081751741
// MI455X (gfx1250) — compile-verified
//
#include <hip/hip_runtime.h>
#include <math.h>

#define EPSF 1e-10f

typedef float v2f __attribute__((ext_vector_type(2)));
typedef float v8f __attribute__((ext_vector_type(8)));
typedef int   i32x4 __attribute__((vector_size(4 * sizeof(int))));  // matches builtin param type

#ifndef __has_builtin
#define __has_builtin(x) 0
#endif

#if __has_builtin(__builtin_amdgcn_global_load_async_to_lds_b128)
#define HAVE_ASYNC_LDS 1
#else
#define HAVE_ASYNC_LDS 0
#endif

// 16-byte global -> LDS copy. Async (ASYNCcnt-tracked, direct-to-LDS) when the
// gfx1250 builtin is available; otherwise plain load/store (compiler-ordered).
__device__ __forceinline__ void copy16_to_lds(const float* __restrict__ g, float* l)
{
#if HAVE_ASYNC_LDS
    __builtin_amdgcn_global_load_async_to_lds_b128(
        (i32x4*)(g), (i32x4*)(l), 0, 0);
#else
    *(float4*)l = *(const float4*)g;
#endif
}

__device__ __forceinline__ void wait_async_lds()
{
#if HAVE_ASYNC_LDS
#if __has_builtin(__builtin_amdgcn_s_wait_asynccnt)
    __builtin_amdgcn_s_wait_asynccnt(0);
#else
    asm volatile("s_wait_asynccnt 0x0" ::: "memory");
#endif
#endif
}

// ---------------------------------------------------------------------------
// Generic fp32 WMMA matmul, fully specialized at compile time:
//   Out[i,n] = act( rowscale[i] * sum_k A[i,k] * B(k,n) )
//   B(k,n) = B[k*ldb + n]  (BTRANS=0)   or   B[n*ldb + k]  (BTRANS=1, e@e^T)
// Block = 256 threads = 8 waves. Block tile: 32 rows x 128 cols.
// Each wave owns a 16-col strip and two 16x16 accumulators (rows 0-15, 16-31),
// iterating v_wmma_f32_16x16x4_f32 over K. A/B tiles double-buffered in LDS:
// async engine fills chunk k+1 while WMMA runs on chunk k.
// Requires: M % 32 == 0, Kdim % 64 == 0, Ncols % 16 == 0.
// ---------------------------------------------------------------------------
template <int BTRANS, int ACT, int HAS_RS>
__global__ __launch_bounds__(256)
void wmma_mm(const float* __restrict__ A, int lda,
             const float* __restrict__ B, int ldb,
             float* __restrict__ Out, int ldo,
             int Kdim, int Ncols,
             const float* __restrict__ rowscale)
{
    __shared__ float As[2][32][68];                                  // 32 x 64 + pad
    __shared__ float BsL[2][BTRANS ? 128 : 64][BTRANS ? 68 : 132];   // B tile

    const int m0   = blockIdx.x * 32;
    const int nb0  = blockIdx.y * 128;
    const int wave = threadIdx.x >> 5;     // 0..7
    const int lane = threadIdx.x & 31;
    const int wn   = wave * 16;            // wave's n-offset inside block tile
    const int hi   = lane >> 4;            // 0: lanes 0-15, 1: lanes 16-31
    const int lm   = lane & 15;
    const bool wave_active = (nb0 + wn < Ncols);

    v8f acc0 = {0.f, 0.f, 0.f, 0.f, 0.f, 0.f, 0.f, 0.f};
    v8f acc1 = {0.f, 0.f, 0.f, 0.f, 0.f, 0.f, 0.f, 0.f};

    // Cooperative tile fill for K-chunk at k0 into buffer p.
    // Out-of-range B columns are left unwritten: they are only ever read by
    // waves with wave_active == false, which never enter the compute loop.
    auto load_tiles = [&](int p, int k0) {
        // A tile: 32 x 64 (512 float4, 2 per thread)
        #pragma unroll
        for (int i = 0; i < 2; ++i) {
            const int idx = threadIdx.x + i * 256;   // 0..511
            const int r   = idx >> 4;                // 0..31
            const int c4  = (idx & 15) * 4;          // 0..60
            copy16_to_lds(A + (size_t)(m0 + r) * lda + (k0 + c4), &As[p][r][c4]);
        }
        if constexpr (BTRANS) {
            // LDS layout [n][k]: contiguous k-pair per lane at compute time
            #pragma unroll
            for (int i = 0; i < 8; ++i) {
                const int idx = threadIdx.x + i * 256;  // 0..2047
                const int n   = idx >> 4;               // 0..127
                const int c4  = (idx & 15) * 4;         // 0..60
                if (nb0 + n < Ncols)
                    copy16_to_lds(B + (size_t)(nb0 + n) * ldb + (k0 + c4), &BsL[p][n][c4]);
            }
        } else {
            // LDS layout [k][n]
            #pragma unroll
            for (int i = 0; i < 8; ++i) {
                const int idx = threadIdx.x + i * 256;  // 0..2047
                const int kr  = idx >> 5;               // 0..63
                const int c4  = (idx & 31) * 4;         // 0..124
                if (nb0 + c4 < Ncols)
                    copy16_to_lds(B + (size_t)(k0 + kr) * ldb + (nb0 + c4), &BsL[p][kr][c4]);
            }
        }
    };

    load_tiles(0, 0);
    wait_async_lds();
    __syncthreads();

    int p = 0;
    for (int k0 = 0; k0 < Kdim; k0 += 64) {
        const bool more = (k0 + 64) < Kdim;
        if (more) load_tiles(p ^ 1, k0 + 64);   // async fill next chunk

        if (wave_active) {
            #pragma unroll
            for (int kk = 0; kk < 64; kk += 4) {
                const int ks = kk + 2 * hi;   // this lane's k-pair within chunk
                v2f a0, a1, b;
                a0.x = As[p][lm][ks];        a0.y = As[p][lm][ks + 1];
                a1.x = As[p][16 + lm][ks];   a1.y = As[p][16 + lm][ks + 1];
                if constexpr (BTRANS) {
                    b.x = BsL[p][wn + lm][ks];
                    b.y = BsL[p][wn + lm][ks + 1];
                } else {
                    b.x = BsL[p][ks][wn + lm];
                    b.y = BsL[p][ks + 1][wn + lm];
                }
                acc0 = __builtin_amdgcn_wmma_f32_16x16x4_f32(
                           false, a0, false, b, (short)0, acc0, false, false);
                acc1 = __builtin_amdgcn_wmma_f32_16x16x4_f32(
                           false, a1, false, b, (short)0, acc1, false, false);
            }
        }

        if (more) wait_async_lds();   // next chunk committed to LDS
        __syncthreads();              // all waves done reading buffer p
        p ^= 1;
    }

    if (wave_active) {
        const int col = nb0 + wn + lm;
        #pragma unroll
        for (int r = 0; r < 8; ++r) {
            const int row0 = m0 + r + 8 * hi;       // acc0: rows 0-15 of tile
            const int row1 = row0 + 16;             // acc1: rows 16-31
            float v0 = acc0[r];
            float v1 = acc1[r];
            if constexpr (HAS_RS) { v0 *= rowscale[row0]; v1 *= rowscale[row1]; }
            if constexpr (ACT == 1) { v0 = tanhf(v0);      v1 = tanhf(v1); }
            if constexpr (ACT == 2) { v0 = fmaxf(v0, 0.f); v1 = fmaxf(v1, 0.f); }
            Out[(size_t)row0 * ldo + col] = v0;
            Out[(size_t)row1 * ldo + col] = v1;
        }
    }
}

// ---------------------------------------------------------------------------
// Row sum of a dense [N x N] matrix. mode==1: write dinv = d>0 ? rsqrt(max(d,eps)) : 0
// mode==0: write raw sum. Deterministic strided reduction (no atomics).
// ---------------------------------------------------------------------------
__global__ __launch_bounds__(256)
void rowsum_kernel(const float* __restrict__ A, int N, float* __restrict__ out, int mode)
{
    const int i = blockIdx.x;
    const float* row = A + (size_t)i * N;
    float s = 0.f;
    for (int j = threadIdx.x; j < N; j += 256) s += row[j];
    __shared__ float red[256];
    red[threadIdx.x] = s;
    __syncthreads();
    for (int st = 128; st > 0; st >>= 1) {
        if (threadIdx.x < st) red[threadIdx.x] += red[threadIdx.x + st];
        __syncthreads();
    }
    if (threadIdx.x == 0) {
        const float d = red[0];
        out[i] = (mode == 1) ? ((d > 0.f) ? rsqrtf(fmaxf(d, EPSF)) : 0.f) : d;
    }
}

// dinv1 / rowscale1 from raw row sums of s (pre-normalization):
//   normalized row-sum: d = rs/(rs+eps); dinv1 = d>0 ? rsqrt(max(d,eps)) : 0
//   rowscale1 = dinv1/(rs+eps)   (folds row normalization into the matmul)
__global__ void dinv1_kernel(const float* __restrict__ rs,
                             float* __restrict__ dinv1, float* __restrict__ rsc1, int N)
{
    const int i = blockIdx.x * blockDim.x + threadIdx.x;
    if (i >= N) return;
    const float r = rs[i];
    const float denom = r + EPSF;
    const float d = r / denom;
    const float dv = (d > 0.f) ? rsqrtf(fmaxf(d, EPSF)) : 0.f;
    dinv1[i] = dv;
    rsc1[i]  = dv / denom;
}

// out[i,f] = in[i,f] * (colw ? colw[f] : 1) * rowsc[i]
__global__ void scale_rows_kernel(const float* __restrict__ in,
                                  const float* __restrict__ colw,
                                  const float* __restrict__ rowsc,
                                  float* __restrict__ out, int rows, int cols)
{
    const size_t t = (size_t)blockIdx.x * blockDim.x + threadIdx.x;
    if (t >= (size_t)rows * cols) return;
    const int i = (int)(t / cols);
    const int f = (int)(t % cols);
    float v = in[t];
    if (colw) v *= colw[f];
    out[t] = v * rowsc[i];
}

// e[i,:] /= (||e[i,:]||_2 + eps), F == blockDim.x == 256
__global__ __launch_bounds__(256)
void rownorm_kernel(float* __restrict__ e, int N, int F)
{
    const int i = blockIdx.x;
    float x = e[(size_t)i * F + threadIdx.x];
    __shared__ float red[256];
    red[threadIdx.x] = x * x;
    __syncthreads();
    for (int st = 128; st > 0; st >>= 1) {
        if (threadIdx.x < st) red[threadIdx.x] += red[threadIdx.x + st];
        __syncthreads();
    }
    const float scale = 1.f / (sqrtf(red[0]) + EPSF);
    e[(size_t)i * F + threadIdx.x] = x * scale;
}

// ---------------------------------------------------------------------------
// Exact top-K (K=32) per row of S[N x N], N = 8192 = 32 * 256.
// Row held in registers (32 f32/thread). Selection is lexicographic
// (value desc, index asc), matching jax.lax.top_k tie behavior.
// ---------------------------------------------------------------------------
__global__ __launch_bounds__(256)
void topk_kernel(const float* __restrict__ S, int N, int K,
                 float* __restrict__ vals, int* __restrict__ idxs)
{
    const int rowi = blockIdx.x;
    const float* rowp = S + (size_t)rowi * N;
    float v[32];
    #pragma unroll
    for (int q = 0; q < 32; ++q) v[q] = rowp[q * 256 + threadIdx.x];

    __shared__ float sval[256];
    __shared__ int   sidx[256];

    float pv = INFINITY;  // previously selected (value, index)
    int   pi = -1;

    for (int t = 0; t < K; ++t) {
        float bv = -INFINITY; int bi = N;
        #pragma unroll
        for (int q = 0; q < 32; ++q) {
            const int j = q * 256 + threadIdx.x;
            const float x = v[q];
            const bool allowed = (x < pv) || (x == pv && j > pi);
            if (allowed && (x > bv || (x == bv && j < bi))) { bv = x; bi = j; }
        }
        sval[threadIdx.x] = bv; sidx[threadIdx.x] = bi;
        __syncthreads();
        for (int st = 128; st > 0; st >>= 1) {
            if (threadIdx.x < st) {
                const float ov = sval[threadIdx.x + st]; const int oi = sidx[threadIdx.x + st];
                const float mv = sval[threadIdx.x];      const int mi = sidx[threadIdx.x];
                if (ov > mv || (ov == mv && oi < mi)) { sval[threadIdx.x] = ov; sidx[threadIdx.x] = oi; }
            }
            __syncthreads();
        }
        pv = sval[0]; pi = sidx[0];
        if (threadIdx.x == 0) {
            vals[(size_t)rowi * K + t] = pv;
            idxs[(size_t)rowi * K + t] = pi;
        }
        __syncthreads();
    }
}

// S (zeroed) <- scatter top-k values: S[i, idx[i,t]] = val[i,t]
__global__ void scatter_kernel(const float* __restrict__ vals, const int* __restrict__ idxs,
                               float* __restrict__ S, int N, int K)
{
    const int t = blockIdx.x * blockDim.x + threadIdx.x;
    if (t >= N * K) return;
    const int i = t / K;
    S[(size_t)i * N + idxs[t]] = vals[t];
}

// In-place: S <- (0.5*(S + S^T)) .* Q, processed as 16x16 tile pairs (bi<=bj).
__global__ __launch_bounds__(256)
void sym_q_kernel(float* __restrict__ S, const float* __restrict__ Q, int N)
{
    const int bi = blockIdx.x, bj = blockIdx.y;
    if (bj < bi) return;
    const int ty = threadIdx.x >> 4, tx = threadIdx.x & 15;
    __shared__ float T1[16][17], T2[16][17];
    const int i1 = bi * 16 + ty, j1 = bj * 16 + tx;
    const int i2 = bj * 16 + ty, j2 = bi * 16 + tx;
    T1[ty][tx] = S[(size_t)i1 * N + j1];
    T2[ty][tx] = S[(size_t)i2 * N + j2];
    __syncthreads();
    const float a1 = 0.5f * (T1[ty][tx] + T2[tx][ty]) * Q[(size_t)i1 * N + j1];
    float a2 = 0.f;
    if (bi != bj) a2 = 0.5f * (T2[ty][tx] + T1[tx][ty]) * Q[(size_t)i2 * N + j2];
    S[(size_t)i1 * N + j1] = a1;
    if (bi != bj) S[(size_t)i2 * N + j2] = a2;
}

// ---------------------------------------------------------------------------
extern "C" void kernel_launch(void* const* d_in, const int* in_sizes, int n_in,
                              void* d_out, int out_size, void* d_ws, size_t ws_size,
                              hipStream_t stream)
{
    const int N = 8192, F = 256, H = 256, C = 32, K = 32;

    const float* x   = (const float*)d_in[0];
    const float* Adj = (const float*)d_in[1];
    const float* Q   = (const float*)d_in[2];
    const float* w1  = (const float*)d_in[3];
    const float* w2  = (const float*)d_in[4];
    const float* W1  = (const float*)d_in[5];
    const float* W2  = (const float*)d_in[6];
    float* out = (float*)d_out;

    char* ws = (char*)d_ws;
    size_t off = 0;
    float* S    = (float*)(ws + off); off += (size_t)N * N * 4;   // 256 MB (sim, then new graph)
    float* e    = (float*)(ws + off); off += (size_t)N * F * 4;   // e, later z
    float* h    = (float*)(ws + off); off += (size_t)N * F * 4;   // h, later Z1/Z2
    float* Bs   = (float*)(ws + off); off += (size_t)N * F * 4;   // pre-scaled B operand
    float* vals = (float*)(ws + off); off += (size_t)N * K * 4;
    int*   idxs = (int*)  (ws + off); off += (size_t)N * K * 4;
    float* dinv0= (float*)(ws + off); off += (size_t)N * 4;
    float* rs1  = (float*)(ws + off); off += (size_t)N * 4;
    float* dinv1= (float*)(ws + off); off += (size_t)N * 4;
    float* rsc1 = (float*)(ws + off); off += (size_t)N * 4;

    const dim3 blk(256);
    const int MB = N / 32;  // 256 row-blocks

    // 1) dinv0 from Adj row degrees
    rowsum_kernel<<<N, blk, 0, stream>>>(Adj, N, dinv0, 1);

    // 2) h = tanh(dinv0_i * Adj @ (dinv0 .* x .* w1))
    scale_rows_kernel<<<(N * F + 255) / 256, blk, 0, stream>>>(x, w1, dinv0, Bs, N, F);
    wmma_mm<0, 1, 1><<<dim3(MB, F / 128), blk, 0, stream>>>(Adj, N, Bs, F, h, F, N, F, dinv0);

    // 3) e = dinv0_i * Adj @ (dinv0 .* h .* w2); row-L2-normalize
    scale_rows_kernel<<<(N * F + 255) / 256, blk, 0, stream>>>(h, w2, dinv0, Bs, N, F);
    wmma_mm<0, 0, 1><<<dim3(MB, F / 128), blk, 0, stream>>>(Adj, N, Bs, F, e, F, N, F, dinv0);
    rownorm_kernel<<<N, blk, 0, stream>>>(e, N, F);

    // 4) sim = e @ e^T  (into S)
    wmma_mm<1, 0, 0><<<dim3(MB, N / 128), blk, 0, stream>>>(e, F, e, F, S, N, F, N, nullptr);

    // 5) top-K per row, then rebuild S as the sparse new graph
    topk_kernel<<<N, blk, 0, stream>>>(S, N, K, vals, idxs);
    (void)hipMemsetAsync(S, 0, (size_t)N * N * 4, stream);
    scatter_kernel<<<(N * K + 255) / 256, blk, 0, stream>>>(vals, idxs, S, N, K);

    // 6) S <- 0.5*(S+S^T) .* Q ; row sums -> rs1 ; derived scales
    sym_q_kernel<<<dim3(N / 16, N / 16), blk, 0, stream>>>(S, Q, N);
    rowsum_kernel<<<N, blk, 0, stream>>>(S, N, rs1, 0);
    dinv1_kernel<<<(N + 255) / 256, blk, 0, stream>>>(rs1, dinv1, rsc1, N);

    // 7) Z1 = x @ W1 (into h);  z = relu(rsc1_i * S @ (dinv1 .* Z1)) (into e)
    wmma_mm<0, 0, 0><<<dim3(MB, H / 128), blk, 0, stream>>>(x, F, W1, H, h, H, F, H, nullptr);
    scale_rows_kernel<<<(N * H + 255) / 256, blk, 0, stream>>>(h, nullptr, dinv1, Bs, N, H);
    wmma_mm<0, 2, 1><<<dim3(MB, H / 128), blk, 0, stream>>>(S, N, Bs, H, e, H, N, H, rsc1);

    // 8) Z2 = z @ W2 (into h);  out = rsc1_i * S @ (dinv1 .* Z2)
    wmma_mm<0, 0, 0><<<dim3(MB, 1), blk, 0, stream>>>(e, H, W2, C, h, C, H, C, nullptr);
    scale_rows_kernel<<<(N * C + 255) / 256, blk, 0, stream>>>(h, nullptr, dinv1, Bs, N, C);
    wmma_mm<0, 0, 1><<<dim3(MB, 1), blk, 0, stream>>>(S, N, Bs, C, out, C, N, C, rsc1);
}